// SingleCellClustering_15006615733369
// MI455X (gfx1250) — compile-verified
//
#include <hip/hip_runtime.h>
#include <float.h>

typedef float v2f __attribute__((ext_vector_type(2)));
typedef float v8f __attribute__((ext_vector_type(8)));

#define KC 50
#define DIM 64
#define NPB 512          // points per block
#define THREADS 256      // 8 waves, each wave does 4 tiles of 16 points
#define CSTRIDE 68       // padded LDS row stride (floats) -> conflict-free b64 reads

// ---------------------------------------------------------------------------
// Assignment + per-block scatter-mean accumulation.
// One wave computes a 16(points) x 64(padded clusters) score tile via
// V_WMMA_F32_16X16X4_F32 chains, argmins it, then the block accumulates
// per-cluster sums/counts in LDS and flushes with global float atomics.
// ---------------------------------------------------------------------------
__global__ __launch_bounds__(THREADS) void kmeans_assign(
    const float* __restrict__ X, const float* __restrict__ C,
    float* __restrict__ gsums, float* __restrict__ gcounts,
    int* __restrict__ labels, int n)
{
    __shared__ float smemC[64 * CSTRIDE];   // centroids (padded), later reused as sums
    __shared__ float csq[64];
    __shared__ int   slab[NPB];
    __shared__ float scnt[KC];

    const int tid  = threadIdx.x;
    const int lane = tid & 31;
    const int wave = tid >> 5;
    const int l15  = lane & 15;
    const int koff = (lane < 16) ? 0 : 2;   // K-pair per lane half (ISA 16x4 f32 layout)
    const int base = blockIdx.x * NPB;

    // Stage centroids into LDS; pad rows 50..63 with zeros.
    for (int i = tid; i < 64 * 64; i += THREADS) {
        int c = i >> 6, d = i & 63;
        smemC[c * CSTRIDE + d] = (c < KC) ? C[c * DIM + d] : 0.0f;
    }
    __syncthreads();
    if (tid < 64) {
        float s = 0.0f;
        for (int d = 0; d < DIM; ++d) { float v = smemC[tid * CSTRIDE + d]; s += v * v; }
        csq[tid] = (tid < KC) ? s : FLT_MAX;   // padded clusters can never win argmin
    }
    __syncthreads();

    for (int t = 0; t < 4; ++t) {
        const int m0  = base + wave * 64 + t * 16;
        int row = m0 + l15;
        if (row >= n) row = n - 1;             // clamp (tail rows are never committed)

        // Preload A fragments: X[row][k..k+1] for this lane's K-pairs, reused 4x.
        const float* xr = X + (size_t)row * DIM + koff;
        v2f a[16];
        #pragma unroll
        for (int j = 0; j < 16; ++j) a[j] = *(const v2f*)(xr + 4 * j);

        float best[8]; int bestn[8];
        #pragma unroll
        for (int r = 0; r < 8; ++r) { best[r] = FLT_MAX; bestn[r] = 0; }

        for (int nt = 0; nt < 4; ++nt) {
            v8f acc = {0.f, 0.f, 0.f, 0.f, 0.f, 0.f, 0.f, 0.f};
            const float* cb = &smemC[(nt * 16 + l15) * CSTRIDE + koff];
            #pragma unroll
            for (int j = 0; j < 16; ++j) {
                v2f b = *(const v2f*)(cb + 4 * j);
                acc = __builtin_amdgcn_wmma_f32_16x16x4_f32(
                        false, a[j], false, b, (short)0, acc, false, false);
            }
            const int   nidx = nt * 16 + l15;
            const float cs   = csq[nidx];
            #pragma unroll
            for (int r = 0; r < 8; ++r) {
                float s = cs - 2.0f * acc[r];          // x^2 term dropped (row-constant)
                if (s < best[r]) { best[r] = s; bestn[r] = nidx; }
            }
        }

        // Cross-lane argmin over the 16 lanes of each half (C/D: N = lane%16).
        #pragma unroll
        for (int mask = 1; mask < 16; mask <<= 1) {
            #pragma unroll
            for (int r = 0; r < 8; ++r) {
                float ob = __shfl_xor(best[r],  mask, 32);
                int   on = __shfl_xor(bestn[r], mask, 32);
                if (ob < best[r] || (ob == best[r] && on < bestn[r])) {
                    best[r] = ob; bestn[r] = on;
                }
            }
        }
        // VGPR r holds row M=r (lanes 0-15) / M=r+8 (lanes 16-31).
        const int rh = (lane < 16) ? 0 : 8;
        if (l15 == 0) {
            #pragma unroll
            for (int r = 0; r < 8; ++r) {
                const int m   = m0 + rh + r;
                const int lab = bestn[r];
                slab[wave * 64 + t * 16 + rh + r] = lab;
                if (m < n) labels[m] = lab;
            }
        }
    }
    __syncthreads();

    // Stage 2: per-block scatter-mean into LDS (reuse centroid region as sums).
    float* ssum = smemC;
    for (int i = tid; i < KC * DIM; i += THREADS) ssum[i] = 0.0f;
    for (int i = tid; i < KC;       i += THREADS) scnt[i] = 0.0f;
    __syncthreads();

    int npts = n - base; if (npts > NPB) npts = NPB;
    for (int idx = tid; idx < npts * DIM; idx += THREADS) {
        const int p = idx >> 6, d = idx & 63;
        atomicAdd(&ssum[slab[p] * DIM + d], X[(size_t)(base + p) * DIM + d]);
    }
    for (int p = tid; p < npts; p += THREADS)
        atomicAdd(&scnt[slab[p]], 1.0f);
    __syncthreads();

    for (int i = tid; i < KC * DIM; i += THREADS) atomicAdd(&gsums[i],   ssum[i]);
    for (int i = tid; i < KC;       i += THREADS) atomicAdd(&gcounts[i], scnt[i]);
}

// ---------------------------------------------------------------------------
// Centroid update: C = counts>0 ? sums/counts : C ; then re-zero accumulators.
// Single block so the counts reads / zeroing are safely ordered.
// ---------------------------------------------------------------------------
__global__ __launch_bounds__(256) void kmeans_update(
    float* __restrict__ C, float* __restrict__ sums, float* __restrict__ counts)
{
    __shared__ float sc[KC];
    const int tid = threadIdx.x;
    if (tid < KC) sc[tid] = counts[tid];
    __syncthreads();
    for (int idx = tid; idx < KC * DIM; idx += 256) {
        const float cnt = sc[idx >> 6];
        const float s   = sums[idx];
        if (cnt > 0.0f) C[idx] = s / cnt;
        sums[idx] = 0.0f;
    }
    if (tid < KC) counts[tid] = 0.0f;
}

// Init: centroids = first K points; zero accumulators.
__global__ __launch_bounds__(256) void kmeans_init(
    const float* __restrict__ X, float* __restrict__ C,
    float* __restrict__ sums, float* __restrict__ counts)
{
    const int idx = blockIdx.x * blockDim.x + threadIdx.x;
    if (idx < KC * DIM) { C[idx] = X[idx]; sums[idx] = 0.0f; }
    if (idx < KC) counts[idx] = 0.0f;
}

extern "C" void kernel_launch(void* const* d_in, const int* in_sizes, int n_in,
                              void* d_out, int out_size, void* d_ws, size_t ws_size,
                              hipStream_t stream)
{
    const float* X = (const float*)d_in[0];
    const int n = in_sizes[0] / DIM;          // 200000

    int*   labels = (int*)d_out;
    float* C      = (float*)d_ws;             // 50*64 f32
    float* sums   = C + KC * DIM;             // 50*64 f32
    float* counts = sums + KC * DIM;          // 50   f32

    kmeans_init<<<(KC * DIM + 255) / 256, 256, 0, stream>>>(X, C, sums, counts);

    const int grid = (n + NPB - 1) / NPB;     // 391 blocks
    for (int it = 0; it < 50; ++it) {
        kmeans_assign<<<grid, THREADS, 0, stream>>>(X, C, sums, counts, labels, n);
        kmeans_update<<<1, 256, 0, stream>>>(C, sums, counts);
    }
}